// SimpleSphereNetModel_37220186587494
// MI455X (gfx1250) — compile-verified
//
#include <hip/hip_runtime.h>
#include <hip/hip_bf16.h>
#include <math.h>

// Problem constants from the reference
#define BB     8
#define NN     80
#define EMBED  512
#define RBF_DIM 64
#define ANGLE_DIM 16
#define LAYERS 4
#define NODES  (BB * NN)          // 640
#define XTOT   (NODES * EMBED)    // 327680

typedef __attribute__((ext_vector_type(2))) float v2f;
typedef __attribute__((ext_vector_type(8))) float v8f;

// ---------------------------------------------------------------------------
// x0[b,n,d] = embed[token[b,n], d] * (token != PAD)
// ---------------------------------------------------------------------------
__global__ void embed_kernel(const int* __restrict__ tokens,
                             const float* __restrict__ emb,
                             float* __restrict__ x)
{
    int idx = blockIdx.x * blockDim.x + threadIdx.x;
    if (idx >= XTOT) return;
    int nd = idx >> 9;          // / EMBED
    int d  = idx & (EMBED - 1);
    int tok = tokens[nd];
    x[idx] = (tok != 0) ? emb[(size_t)tok * EMBED + d] : 0.0f;
}

// ---------------------------------------------------------------------------
// Per (b,i): ef_agg[0:64]  = sum_j adj_ij * exp(-10*(dist_ij - c_t)^2)
//            ef_agg[64:80] = (sum_{j!=k, adj_ij & adj_ik} silu(cos*w1+b1)) @ w2
//                            + cnt * b2
//            deg[b,i]      = sum_j adj_ij
// One block per node, 128 threads.
// ---------------------------------------------------------------------------
__global__ __launch_bounds__(128) void edge_agg_kernel(
    const int* __restrict__ tokens, const float* __restrict__ coord,
    const float* __restrict__ w1, const float* __restrict__ b1,
    const float* __restrict__ w2, const float* __restrict__ b2,
    float* __restrict__ ef_agg, float* __restrict__ deg)
{
    const int node = blockIdx.x;            // b*NN + i
    const int b    = node / NN;
    const int i    = node % NN;
    const int tid  = threadIdx.x;

    __shared__ float svx[NN], svy[NN], svz[NN], sdist[NN], sinv[NN], sadj[NN];
    __shared__ float sh[ANGLE_DIM];
    __shared__ float scnt;
    __shared__ float cix, ciy, ciz;
    __shared__ int   validi;

    if (tid == 0) {
        validi = (tokens[b * NN + i] != 0);
        const float* ci = coord + (size_t)(b * NN + i) * 3;
        cix = ci[0]; ciy = ci[1]; ciz = ci[2];
        scnt = 0.0f;
    }
    if (tid < ANGLE_DIM) sh[tid] = 0.0f;
    __syncthreads();

    if (tid < NN) {
        int j = tid;
        const float* cj = coord + (size_t)(b * NN + j) * 3;
        float dx = cix - cj[0], dy = ciy - cj[1], dz = ciz - cj[2]; // v_ij = c_i - c_j
        float sq = dx * dx + dy * dy + dz * dz;
        float dist = (sq > 0.0f) ? sqrtf(sq) : 0.0f;
        int vj = (tokens[b * NN + j] != 0);
        float adj = (validi && vj && (j != i) && (dist <= 6.0f)) ? 1.0f : 0.0f;
        svx[j] = dx; svy[j] = dy; svz[j] = dz;
        sdist[j] = dist;
        sinv[j] = 1.0f / fmaxf(dist, 1e-8f);
        sadj[j] = adj;
    }
    __syncthreads();

    // RBF aggregation: thread t owns center c_t = 6*t/63
    if (tid < RBF_DIM) {
        float c = 6.0f * (float)tid / (float)(RBF_DIM - 1);
        float acc = 0.0f;
        for (int j = 0; j < NN; ++j) {
            float dlt = sdist[j] - c;
            acc += sadj[j] * __expf(-10.0f * dlt * dlt);
        }
        ef_agg[(size_t)node * (RBF_DIM + ANGLE_DIM) + tid] = acc;
    }

    // Triplet-angle aggregation over (j,k) pairs
    float w1r[ANGLE_DIM], b1r[ANGLE_DIM], hacc[ANGLE_DIM];
#pragma unroll
    for (int a = 0; a < ANGLE_DIM; ++a) { w1r[a] = w1[a]; b1r[a] = b1[a]; hacc[a] = 0.0f; }
    float cnt = 0.0f;

    for (int p = tid; p < NN * NN; p += blockDim.x) {
        int j = p / NN, k = p - j * NN;
        if (j == k) continue;
        if (sadj[j] == 0.0f || sadj[k] == 0.0f) continue;
        float dot = svx[j] * svx[k] + svy[j] * svy[k] + svz[j] * svz[k];
        float cosv = dot * sinv[j] * sinv[k];
        cosv = fminf(1.0f, fmaxf(-1.0f, cosv));
        cnt += 1.0f;
#pragma unroll
        for (int a = 0; a < ANGLE_DIM; ++a) {
            float z = cosv * w1r[a] + b1r[a];
            hacc[a] += z / (1.0f + __expf(-z));       // silu
        }
    }
#pragma unroll
    for (int a = 0; a < ANGLE_DIM; ++a) atomicAdd(&sh[a], hacc[a]);
    atomicAdd(&scnt, cnt);
    __syncthreads();

    if (tid < ANGLE_DIM) {
        float acc = scnt * b2[tid];
        for (int a1 = 0; a1 < ANGLE_DIM; ++a1) acc += sh[a1] * w2[a1 * ANGLE_DIM + tid];
        ef_agg[(size_t)node * (RBF_DIM + ANGLE_DIM) + RBF_DIM + tid] = acc;
    }
    if (tid == 0) {
        float dsum = 0.0f;
        for (int j = 0; j < NN; ++j) dsum += sadj[j];
        deg[node] = dsum;
    }
}

// ---------------------------------------------------------------------------
// Layer-batched GEMM: for layer l = blockIdx.z,
//   C_l[M,N] = op( A_l[M,K] @ B_l[K,N] + bias_l[N]*(rowScale?rowScale[m]:1) )
//                  [+ Cadd_l]
// flags: bit0 = ReLU, bit1 = add Cadd.
// One wave32 computes one 16x16 tile via V_WMMA_F32_16X16X4_F32, K step 4.
// The 16-row A panel is staged once per block in LDS (row pad +4 dwords ->
// the 16 lanes of a half touch 32 distinct banks on the b64 fragment reads).
// Fragment layouts per CDNA5 ISA 7.12.2; M,N,K divide tile sizes evenly so
// EXEC is all ones at every WMMA.
// ---------------------------------------------------------------------------
__global__ __launch_bounds__(256) void gemm_wmma_f32(
    const float* __restrict__ A,  long long aStride,
    const float* __restrict__ Bw, long long bStride,
    const float* __restrict__ bias, long long biasStride,
    const float* __restrict__ rowScale,
    float* __restrict__ C, long long cStride,
    const float* __restrict__ Cadd,
    int M, int K, int N, int flags)
{
    const int l = blockIdx.z;
    A  += (size_t)l * aStride;
    Bw += (size_t)l * bStride;
    C  += (size_t)l * cStride;
    if (bias) bias += (size_t)l * biasStride;
    if (Cadd) Cadd += (size_t)l * cStride;

    extern __shared__ float As[];           // [16][K+4]
    const int KP = K + 4;
    const int m0 = blockIdx.x * 16;

    // Stage the 16 x K A panel (shared by all 8 waves of this block)
    for (int idx = threadIdx.x; idx < 16 * K; idx += 256) {
        int rr = idx / K, cc = idx - rr * K;
        As[rr * KP + cc] = A[(size_t)(m0 + rr) * K + cc];
    }
    __syncthreads();

    const int wave = threadIdx.x >> 5;
    const int lane = threadIdx.x & 31;
    const int n0 = blockIdx.y * 128 + wave * 16;
    const int h = lane >> 4;                // K/M half select
    const int r = lane & 15;

    v8f acc = {};
    const float* asrc = As + r * KP;
    for (int k0 = 0; k0 < K; k0 += 4) {
        int kk = k0 + 2 * h;
        v2f a, bfr;
        a.x   = asrc[kk];
        a.y   = asrc[kk + 1];
        bfr.x = Bw[(size_t)kk * N + n0 + r];
        bfr.y = Bw[(size_t)(kk + 1) * N + n0 + r];
        acc = __builtin_amdgcn_wmma_f32_16x16x4_f32(
            /*neg_a=*/false, a, /*neg_b=*/false, bfr,
            /*c_mod=*/(short)0, acc, /*reuse_a=*/false, /*reuse_b=*/false);
    }

    float bn = bias ? bias[n0 + r] : 0.0f;
#pragma unroll
    for (int v = 0; v < 8; ++v) {
        int m = m0 + v + 8 * h;
        float scale = rowScale ? rowScale[m] : 1.0f;
        float val = acc[v] + bn * scale;
        if (flags & 1) val = fmaxf(val, 0.0f);
        size_t idx = (size_t)m * N + (n0 + r);
        if (flags & 2) val += Cadd[idx];
        C[idx] = val;
    }
}

// ---------------------------------------------------------------------------
// x = (x + sum_l contrib_l) * vf ; emit padding mask (bool -> 0/1 float)
// ---------------------------------------------------------------------------
__global__ void finalize_kernel(const int* __restrict__ tokens,
                                float* __restrict__ x,
                                float* __restrict__ maskout,
                                const float* __restrict__ contrib, int nsum)
{
    int idx = blockIdx.x * blockDim.x + threadIdx.x;
    if (idx < XTOT) {
        int nd = idx >> 9;
        float v = x[idx];
        for (int l = 0; l < nsum; ++l) v += contrib[(size_t)l * XTOT + idx];
        x[idx] = (tokens[nd] != 0) ? v : 0.0f;
    }
    if (idx < NODES) maskout[idx] = (tokens[idx] == 0) ? 1.0f : 0.0f;
}

// ---------------------------------------------------------------------------
extern "C" void kernel_launch(void* const* d_in, const int* in_sizes, int n_in,
                              void* d_out, int out_size, void* d_ws, size_t ws_size,
                              hipStream_t stream)
{
    const int*   tokens = (const int*)  d_in[0];
    const float* coord  = (const float*)d_in[1];
    // d_in[2] src_distance, d_in[3] src_edge_type: unused by the math
    const float* emb    = (const float*)d_in[4];
    const float* aw1    = (const float*)d_in[5];
    const float* ab1    = (const float*)d_in[6];
    const float* aw2    = (const float*)d_in[7];
    const float* ab2    = (const float*)d_in[8];
    const float* ew     = (const float*)d_in[9];
    const float* eb     = (const float*)d_in[10];
    const float* nw1    = (const float*)d_in[11];
    const float* nb1    = (const float*)d_in[12];
    const float* nw2    = (const float*)d_in[13];
    const float* nb2    = (const float*)d_in[14];

    float* x    = (float*)d_out;          // [640, 512] accumulator = encoder_rep
    float* mask = x + XTOT;               // [640] padding mask (as 0/1 floats)

    float* ws     = (float*)d_ws;
    float* ef_agg = ws;                   // 640*80 = 51200
    float* deg    = ws + 51200;           // 640
    const size_t base = 52224;            // aligned start of GEMM scratch

    const size_t shK80  = 16 * (80  + 4) * sizeof(float);
    const size_t shK512 = 16 * (512 + 4) * sizeof(float);

    // 1) token embedding (masked)
    embed_kernel<<<(XTOT + 255) / 256, 256, 0, stream>>>(tokens, emb, x);

    // 2) fused edge-feature aggregation (RBF + triplet angles), once for all layers
    edge_agg_kernel<<<NODES, 128, 0, stream>>>(tokens, coord, aw1, ab1, aw2, ab2,
                                               ef_agg, deg);

    const size_t needPar = (base + 8ull * XTOT) * sizeof(float);  // ~10.7 MB
    const size_t needSeq = (base + 2ull * XTOT) * sizeof(float);  // ~2.8 MB

    if (ws_size >= needPar) {
        // --- Layer-parallel path: all 4 layers as gridDim.z (640 blocks/GEMM)
        float* t1   = ws + base;            // [4][640*512]; reused as contrib
        float* hmid = t1 + 4ull * XTOT;     // [4][640*512]
        dim3 g(NODES / 16, EMBED / 128, LAYERS);
        // t1_l = ef_agg @ edge_w[l] + deg * edge_b[l]
        gemm_wmma_f32<<<g, 256, shK80, stream>>>(
            ef_agg, 0, ew, 80ll * EMBED, eb, EMBED, deg,
            t1, XTOT, nullptr, NODES, 80, EMBED, 0);
        // hmid_l = relu(t1_l @ node_w1[l] + node_b1[l])
        gemm_wmma_f32<<<g, 256, shK512, stream>>>(
            t1, XTOT, nw1, (long long)EMBED * EMBED, nb1, EMBED, nullptr,
            hmid, XTOT, nullptr, NODES, EMBED, EMBED, 1);
        // contrib_l = hmid_l @ node_w2[l] + node_b2[l]   (into t1, now dead)
        gemm_wmma_f32<<<g, 256, shK512, stream>>>(
            hmid, XTOT, nw2, (long long)EMBED * EMBED, nb2, EMBED, nullptr,
            t1, XTOT, nullptr, NODES, EMBED, EMBED, 0);
        // x = (x0 + sum_l contrib_l) * vf
        finalize_kernel<<<(XTOT + 255) / 256, 256, 0, stream>>>(tokens, x, mask,
                                                                t1, LAYERS);
    } else if (ws_size >= needSeq) {
        // --- Sequential fallback (smaller scratch)
        float* t1   = ws + base;
        float* hmid = t1 + XTOT;
        dim3 g(NODES / 16, EMBED / 128, 1);
        for (int l = 0; l < LAYERS; ++l) {
            gemm_wmma_f32<<<g, 256, shK80, stream>>>(
                ef_agg, 0, ew + (size_t)l * 80 * EMBED, 0, eb + (size_t)l * EMBED, 0,
                deg, t1, 0, nullptr, NODES, 80, EMBED, 0);
            gemm_wmma_f32<<<g, 256, shK512, stream>>>(
                t1, 0, nw1 + (size_t)l * EMBED * EMBED, 0, nb1 + (size_t)l * EMBED, 0,
                nullptr, hmid, 0, nullptr, NODES, EMBED, EMBED, 1);
            gemm_wmma_f32<<<g, 256, shK512, stream>>>(
                hmid, 0, nw2 + (size_t)l * EMBED * EMBED, 0, nb2 + (size_t)l * EMBED, 0,
                nullptr, x, 0, x, NODES, EMBED, EMBED, 2);
        }
        finalize_kernel<<<(XTOT + 255) / 256, 256, 0, stream>>>(tokens, x, mask,
                                                                nullptr, 0);
    }
}